// NeuralDisCoCirc_19688130085803
// MI455X (gfx1250) — compile-verified
//
#include <hip/hip_runtime.h>
#include <hip/hip_bf16.h>

// Chain of 8 per-sample block GEMV layers:
//   y[b,j] = sum_i x[b,i] * W[l,b,i,j] + bias[l,b,j];  y = mask ? relu(y) : y
// L=8, B=64, D=512. Bandwidth-bound (512 MB of weights, each read once).
//
// v3: split-K=2 per (b, j-tile) -> 4096 waves, LDS reduction; W stream
// addressed as one base pointer + immediate offsets so the inner loop is
// pure clause'd loads + WMMA (no per-load 64-bit address rebuilds).

typedef float v2f __attribute__((ext_vector_type(2)));
typedef float v8f __attribute__((ext_vector_type(8)));

#define NDC_L 8
#define NDC_B 64
#define NDC_D 512

// Block: 128 threads = 4 waves = 2 j-tiles x 2 k-halves for one batch element.
// Each wave: A = W^T tile (16 j-rows x 4 i), B = x replicated into all 16
// columns, C accumulates over its 256-deep i half. All columns of D are
// identical, so lane 0 holds y[j0..j0+7] in c[0..7], lane 16 holds
// y[j0+8..j0+15]. k-half 1 dumps partials to LDS; k-half 0 reduces + epilogue.
__global__ __launch_bounds__(128) void ndc_layer_kernel(
    const float* __restrict__ x,     // [B][D]  layer input
    const float* __restrict__ W,     // [B][D][D] this layer's weights
    const float* __restrict__ bias,  // [B][D]
    const int*   __restrict__ mask,  // [B][D]
    float* __restrict__ y)           // [B][D]  layer output
{
    const int b    = blockIdx.y;
    const int tid  = threadIdx.x;
    const int lane = tid & 31;
    const int wave = tid >> 5;                       // 0..3
    const int t    = wave >> 1;                      // j-tile in block: 0..1
    const int kh   = wave & 1;                       // k-half: 0..1
    const int j0   = blockIdx.x * 32 + t * 16;       // 16-wide j tile per wave

    __shared__ float xs[NDC_D];
    __shared__ float red[2 * 16];                    // split-K partials, per tile

    // Stage x[b][*] in LDS: 128 threads x 1 float4 = 512 floats.
    {
        const float4* src = (const float4*)(x + (size_t)b * NDC_D);
        ((float4*)xs)[tid] = src[tid];
    }
    __syncthreads();

    // A-matrix layout (16x4 f32): lanes 0-15 -> M=0..15, v0 holds K=0 (lanes
    // 0-15) / K=2 (lanes 16-31); v1 holds K=1 / K=3.
    const int ki    = (lane < 16) ? 0 : 2;
    const int jl    = j0 + (lane & 15);
    const int ibase = kh * (NDC_D / 2);              // this wave's i half

    // Single running base pointer; all loads below are wp[imm] so the loop
    // body is clause'd global_load_b32 with immediate offsets.
    const float* wp = W + (size_t)b * NDC_D * NDC_D + (size_t)(ibase + ki) * NDC_D + jl;
    int xi = ibase + ki;

    v8f c = {};                                      // f32 accumulator (8 VGPRs)

#pragma unroll 1
    for (int it = 0; it < (NDC_D / 2) / 16; ++it) {  // 16 iterations x 16 i-rows
        // Prefetch 64 rows (128 KB) ahead into GL2 (global_prefetch_b8).
        __builtin_prefetch(wp + 64 * NDC_D, 0, 3);
        __builtin_prefetch(wp + 65 * NDC_D, 0, 3);
#pragma unroll
        for (int u = 0; u < 4; ++u) {
            v2f a, xb;
            // W^T fragment: 2 dword loads at immediate offsets (u*8KB, +2KB),
            // half-wave contiguous in j.
            a.x = wp[u * 4 * NDC_D];
            a.y = wp[u * 4 * NDC_D + NDC_D];
            // x fragment: replicate the row value across the half-wave so
            // every B column equals x (contiguous pair -> ds_load_b64).
            xb.x = xs[xi + u * 4];
            xb.y = xs[xi + u * 4 + 1];
            c = __builtin_amdgcn_wmma_f32_16x16x4_f32(
                    /*neg_a=*/false, a, /*neg_b=*/false, xb,
                    /*c_mod=*/(short)0, c, /*reuse_a=*/false, /*reuse_b=*/false);
        }
        wp += 16 * NDC_D;                            // one 64-bit add per iter
        xi += 16;
    }

    // Split-K reduction: k-half 1 stores its 16 useful partials to LDS.
    if (kh == 1 && (lane & 15) == 0) {
        float* r = &red[t * 16 + (lane >> 4) * 8];   // lane0 -> +0, lane16 -> +8
#pragma unroll
        for (int r8 = 0; r8 < 8; ++r8) r[r8] = c[r8];
    }
    __syncthreads();

    // k-half 0 reduces and runs the epilogue: lanes 0 and 16 each own 8 j's.
    if (kh == 0 && (lane & 15) == 0) {
        const int off   = (lane >> 4) * 8;           // lane0 -> 0, lane16 -> 8
        const int jbase = j0 + off;
        const float* rp = &red[t * 16 + off];
        const float* bl = bias + (size_t)b * NDC_D + jbase;
        const int*   ml = mask + (size_t)b * NDC_D + jbase;
        float*       yo = y    + (size_t)b * NDC_D + jbase;
#pragma unroll
        for (int r8 = 0; r8 < 8; ++r8) {
            float v = c[r8] + rp[r8] + bl[r8];
            if (ml[r8]) v = fmaxf(v, 0.0f);
            yo[r8] = v;
        }
    }
}

extern "C" void kernel_launch(void* const* d_in, const int* in_sizes, int n_in,
                              void* d_out, int out_size, void* d_ws, size_t ws_size,
                              hipStream_t stream) {
    (void)in_sizes; (void)n_in; (void)out_size; (void)ws_size;

    const float* x       = (const float*)d_in[0];   // [B][D]
    const float* weights = (const float*)d_in[1];   // [L][B][D][D]
    const float* biases  = (const float*)d_in[2];   // [L][B][D]
    const int*   masks   = (const int*)  d_in[3];   // [L][B][D]
    float*       out     = (float*)d_out;           // [B][D]

    // Ping-pong activation buffers in workspace (2 x 128 KB).
    float* buf0 = (float*)d_ws;
    float* buf1 = buf0 + (size_t)NDC_B * NDC_D;

    const dim3 grid(NDC_D / 32, NDC_B, 1);   // 16 x 64 blocks
    const dim3 block(128, 1, 1);             // 4 waves: 2 j-tiles x 2 k-halves

    const size_t wstride = (size_t)NDC_B * NDC_D * NDC_D;
    const size_t vstride = (size_t)NDC_B * NDC_D;

    for (int l = 0; l < NDC_L; ++l) {
        const float* in  = (l == 0) ? x : ((l & 1) ? buf0 : buf1);
        float*       dst = (l == NDC_L - 1) ? out : ((l & 1) ? buf1 : buf0);
        ndc_layer_kernel<<<grid, block, 0, stream>>>(
            in,
            weights + (size_t)l * wstride,
            biases  + (size_t)l * vstride,
            masks   + (size_t)l * vstride,
            dst);
    }
}